// GraphModule_59012850647678
// MI455X (gfx1250) — compile-verified
//
#include <hip/hip_runtime.h>

// CDNA5 / gfx1250. Signed-GCN forward. f32 WMMA (V_WMMA_F32_16X16X4_F32) for
// the dense node transforms; wave-per-edge f32 scatter atomics (L2-resident)
// for the mean aggregations.

typedef float v2f __attribute__((ext_vector_type(2)));
typedef float v8f __attribute__((ext_vector_type(8)));

#define FEAT 32

// Per-lane fragment covering a 16x32 (MxK) operand: 8 chunks of K=4.
// ISA layout (32-bit A 16x4): lane l (l16 = l&15, h = l>>4) holds
//   {M=l16, K=4kc+2h} and {M=l16, K=4kc+2h+1}.
// B (4x16, K x N) uses the mirrored pattern with N = l16 (+16*nt), rows of W.
struct FragK32 { v2f k[8]; };

__device__ __forceinline__ FragK32 load_frag(const float* __restrict__ rowptr,
                                             int half, float scale) {
  FragK32 f;
  const float* p = rowptr + 2 * half;
#pragma unroll
  for (int kc = 0; kc < 8; ++kc) {
    f.k[kc].x = p[4 * kc + 0] * scale;
    f.k[kc].y = p[4 * kc + 1] * scale;
  }
  return f;
}

__device__ __forceinline__ v8f wmma_k32(const FragK32& a, const FragK32& b, v8f c) {
#pragma unroll
  for (int kc = 0; kc < 8; ++kc) {
    // (neg_a, A, neg_b, B, c_mod, C, reuse_a, reuse_b)
    c = __builtin_amdgcn_wmma_f32_16x16x4_f32(false, a.k[kc], false, b.k[kc],
                                              (short)0, c, false, false);
  }
  return c;
}

// ---------------------------------------------------------------------------

__global__ void zero_kernel(float* __restrict__ p, long long n) {
  long long i = (long long)blockIdx.x * blockDim.x + threadIdx.x;
  if (i < n) p[i] = 0.0f;
}

__global__ void count_kernel(const int* __restrict__ dst, float* __restrict__ cnt, int E) {
  int e = blockIdx.x * blockDim.x + threadIdx.x;
  if (e < E) atomicAdd(&cnt[dst[e]], 1.0f);
}

// One wave per edge; 32 lanes <-> 32 features. feat has row stride fstride.
__global__ void scatter_kernel(const float* __restrict__ feat, int fstride,
                               const int* __restrict__ src, const int* __restrict__ dst,
                               float* __restrict__ out, int E) {
  unsigned tid = blockIdx.x * blockDim.x + threadIdx.x;
  int wave = (int)(tid >> 5);
  if (wave >= E) return;
  int lane = threadIdx.x & 31;
  int s = src[wave];
  int d = dst[wave];
  float v = feat[(size_t)s * fstride + lane];
  atomicAdd(out + (size_t)d * FEAT + lane, v);
}

// ---------------------------------------------------------------------------
// Layer 1: z[i, :32] = relu(meanP[i] @ w1pl^T + x[i] @ w1pr^T + b1p)
//          z[i,32:64]= relu(meanN[i] @ w1nl^T + x[i] @ w1nr^T + b1n)
// One wave handles 16 nodes.
__global__ void layer1_kernel(const float* __restrict__ x,
                              const float* __restrict__ sumP, const float* __restrict__ sumN,
                              const float* __restrict__ cntP, const float* __restrict__ cntN,
                              const float* __restrict__ w1pl, const float* __restrict__ w1pr,
                              const float* __restrict__ b1p,
                              const float* __restrict__ w1nl, const float* __restrict__ w1nr,
                              const float* __restrict__ b1n,
                              float* __restrict__ z, int n) {
  unsigned tid = blockIdx.x * blockDim.x + threadIdx.x;
  int wave = (int)(tid >> 5);
  int lane = threadIdx.x & 31;
  int half = lane >> 4, l16 = lane & 15;
  int base = wave * 16;
  if (base >= n) return;                 // uniform per wave
  int row = base + l16; if (row >= n) row = n - 1;

  float invP = 1.0f / fmaxf(cntP[row], 1.0f);
  float invN = 1.0f / fmaxf(cntN[row], 1.0f);

  FragK32 aP = load_frag(sumP + (size_t)row * FEAT, half, invP);
  FragK32 aN = load_frag(sumN + (size_t)row * FEAT, half, invN);
  FragK32 ax = load_frag(x    + (size_t)row * FEAT, half, 1.0f);

#pragma unroll
  for (int nt = 0; nt < 2; ++nt) {
    int col = nt * 16 + l16;

    FragK32 bPl = load_frag(w1pl + (size_t)col * FEAT, half, 1.0f);
    FragK32 bPr = load_frag(w1pr + (size_t)col * FEAT, half, 1.0f);
    v8f accP = {0.f,0.f,0.f,0.f,0.f,0.f,0.f,0.f};
    accP = wmma_k32(aP, bPl, accP);
    accP = wmma_k32(ax, bPr, accP);
    float biasP = b1p[col];
#pragma unroll
    for (int v = 0; v < 8; ++v) {
      int r = base + v + 8 * half;
      float val = fmaxf(accP[v] + biasP, 0.0f);
      if (r < n) z[(size_t)r * 64 + col] = val;
    }

    FragK32 bNl = load_frag(w1nl + (size_t)col * FEAT, half, 1.0f);
    FragK32 bNr = load_frag(w1nr + (size_t)col * FEAT, half, 1.0f);
    v8f accN = {0.f,0.f,0.f,0.f,0.f,0.f,0.f,0.f};
    accN = wmma_k32(aN, bNl, accN);
    accN = wmma_k32(ax, bNr, accN);
    float biasN = b1n[col];
#pragma unroll
    for (int v = 0; v < 8; ++v) {
      int r = base + v + 8 * half;
      float val = fmaxf(accN[v] + biasN, 0.0f);
      if (r < n) z[(size_t)r * 64 + 32 + col] = val;
    }
  }
}

// ---------------------------------------------------------------------------
// Layer 2 (one half): out[i, ooff:ooff+32] =
//   relu([meanA | meanB](i) @ wl^T + z[i, xoff:xoff+32] @ wr^T + bias)
// wl is (32,64) row-major; meanA scaled by 1/cntP, meanB by 1/cntN.
__global__ void layer2_kernel(const float* __restrict__ z, int xoff,
                              const float* __restrict__ sumA, const float* __restrict__ sumB,
                              const float* __restrict__ cntP, const float* __restrict__ cntN,
                              const float* __restrict__ wl, const float* __restrict__ wr,
                              const float* __restrict__ bias,
                              float* __restrict__ out, int ooff, int n) {
  unsigned tid = blockIdx.x * blockDim.x + threadIdx.x;
  int wave = (int)(tid >> 5);
  int lane = threadIdx.x & 31;
  int half = lane >> 4, l16 = lane & 15;
  int base = wave * 16;
  if (base >= n) return;                 // uniform per wave
  int row = base + l16; if (row >= n) row = n - 1;

  float invA = 1.0f / fmaxf(cntP[row], 1.0f);
  float invB = 1.0f / fmaxf(cntN[row], 1.0f);

  FragK32 aA = load_frag(sumA + (size_t)row * FEAT, half, invA);
  FragK32 aB = load_frag(sumB + (size_t)row * FEAT, half, invB);
  FragK32 ax = load_frag(z + (size_t)row * 64 + xoff, half, 1.0f);

#pragma unroll
  for (int nt = 0; nt < 2; ++nt) {
    int col = nt * 16 + l16;

    FragK32 bA = load_frag(wl + (size_t)col * 64,      half, 1.0f); // k in [0,32)
    FragK32 bB = load_frag(wl + (size_t)col * 64 + 32, half, 1.0f); // k in [32,64)
    FragK32 bR = load_frag(wr + (size_t)col * FEAT,    half, 1.0f);

    v8f acc = {0.f,0.f,0.f,0.f,0.f,0.f,0.f,0.f};
    acc = wmma_k32(aA, bA, acc);
    acc = wmma_k32(aB, bB, acc);
    acc = wmma_k32(ax, bR, acc);
    float b = bias[col];
#pragma unroll
    for (int v = 0; v < 8; ++v) {
      int r = base + v + 8 * half;
      float val = fmaxf(acc[v] + b, 0.0f);
      if (r < n) out[(size_t)r * 64 + ooff + col] = val;
    }
  }
}

// ---------------------------------------------------------------------------

extern "C" void kernel_launch(void* const* d_in, const int* in_sizes, int n_in,
                              void* d_out, int out_size, void* d_ws, size_t ws_size,
                              hipStream_t stream) {
  const float* x    = (const float*)d_in[0];
  const int*   pe   = (const int*)d_in[1];   // (2, E): [src | dst]
  const int*   ne   = (const int*)d_in[2];
  const float* w1pl = (const float*)d_in[3];
  const float* w1pr = (const float*)d_in[4];
  const float* b1p  = (const float*)d_in[5];
  const float* w1nl = (const float*)d_in[6];
  const float* w1nr = (const float*)d_in[7];
  const float* b1n  = (const float*)d_in[8];
  const float* w2pl = (const float*)d_in[9];
  const float* w2pr = (const float*)d_in[10];
  const float* b2p  = (const float*)d_in[11];
  const float* w2nl = (const float*)d_in[12];
  const float* w2nr = (const float*)d_in[13];
  const float* b2n  = (const float*)d_in[14];
  float* out = (float*)d_out;

  const int n = in_sizes[0] / FEAT;
  const int E = in_sizes[1] / 2;

  // Workspace layout (floats): cntP[n] cntN[n] bufA[32n] bufB[32n] z[64n]
  float* ws   = (float*)d_ws;
  float* cntP = ws;
  float* cntN = ws + (size_t)n;
  float* bufA = ws + 2 * (size_t)n;
  float* bufB = bufA + 32 * (size_t)n;
  float* z    = bufB + 32 * (size_t)n;

  const int T = 256;
  auto blocks_for = [](long long cnt, int t) -> unsigned {
    return (unsigned)((cnt + t - 1) / t);
  };
  const unsigned edgeWaveBlocks  = (unsigned)(((long long)E * 32 + T - 1) / T);
  const unsigned edgeThreadBlocks = (unsigned)(((long long)E + T - 1) / T);
  const int tiles = (n + 15) / 16;
  const unsigned nodeBlocks = (unsigned)((tiles + 7) / 8); // 8 waves/block

  // 1) zero counts + first-layer sums (66n floats, contiguous)
  zero_kernel<<<blocks_for(66LL * n, T), T, 0, stream>>>(ws, 66LL * n);

  // 2) degrees
  count_kernel<<<edgeThreadBlocks, T, 0, stream>>>(pe + E, cntP, E);
  count_kernel<<<edgeThreadBlocks, T, 0, stream>>>(ne + E, cntN, E);

  // 3) segment sums of x
  scatter_kernel<<<edgeWaveBlocks, T, 0, stream>>>(x, FEAT, pe, pe + E, bufA, E);
  scatter_kernel<<<edgeWaveBlocks, T, 0, stream>>>(x, FEAT, ne, ne + E, bufB, E);

  // 4) layer 1 -> z[N,64]
  layer1_kernel<<<nodeBlocks, T, 0, stream>>>(x, bufA, bufB, cntP, cntN,
                                              w1pl, w1pr, b1p, w1nl, w1nr, b1n,
                                              z, n);

  // 5) out_pos2: aggregates mean(x_p,pos), mean(x_n,neg)
  zero_kernel<<<blocks_for(64LL * n, T), T, 0, stream>>>(bufA, 64LL * n);
  scatter_kernel<<<edgeWaveBlocks, T, 0, stream>>>(z + 0,  64, pe, pe + E, bufA, E);
  scatter_kernel<<<edgeWaveBlocks, T, 0, stream>>>(z + 32, 64, ne, ne + E, bufB, E);
  layer2_kernel<<<nodeBlocks, T, 0, stream>>>(z, 0, bufA, bufB, cntP, cntN,
                                              w2pl, w2pr, b2p, out, 0, n);

  // 6) out_neg2: aggregates mean(x_n,pos), mean(x_p,neg)
  zero_kernel<<<blocks_for(64LL * n, T), T, 0, stream>>>(bufA, 64LL * n);
  scatter_kernel<<<edgeWaveBlocks, T, 0, stream>>>(z + 32, 64, pe, pe + E, bufA, E);
  scatter_kernel<<<edgeWaveBlocks, T, 0, stream>>>(z + 0,  64, ne, ne + E, bufB, E);
  layer2_kernel<<<nodeBlocks, T, 0, stream>>>(z, 32, bufA, bufB, cntP, cntN,
                                              w2nl, w2nr, b2n, out, 32, n);
}